// ContextGating_72292889526407
// MI455X (gfx1250) — compile-verified
//
#include <hip/hip_runtime.h>
#include <cstdint>

// ---------------------------------------------------------------------------
// ContextGating fused kernel for MI455X (gfx1250, wave32)
//   h_norm = rmsnorm(hidden, w_h);  e_norm = rmsnorm(engrams, w_e)
//   scores = <h_norm, e_norm> * 1/sqrt(H);  alpha = sigmoid(scores + bias)
//   contribution = sum_k alpha_k * engrams_k
// Memory-bound (~335 MB @ 23.3 TB/s => ~14 us floor). Engrams are streamed
// from HBM exactly once via async global->LDS DMA (double-buffered rows,
// th:TH_LOAD_NT since the 268 MB stream exceeds the 192 MB L2), consumed
// twice (dot + weighted accumulate) from registers. Hidden rows and the
// contribution output are also single-touch and use non-temporal policy;
// w_h/w_e (16 KB, reused by all 4096 blocks) stay regular-temporal.
// ---------------------------------------------------------------------------

constexpr int   kH       = 2048;
constexpr int   kK       = 8;
constexpr int   kThreads = 256;              // 8 wave32
constexpr int   kEPT     = kH / kThreads;    // 8 floats / thread
constexpr int   kWaves   = kThreads / 32;
constexpr float kEps     = 1e-6f;
constexpr float kScale   = 0.02209708691207961f;  // 1/sqrt(2048)

// Native clang vector (required by __builtin_nontemporal_load/store; HIP's
// float4 is a class type and is rejected by the builtin).
typedef float v4f __attribute__((ext_vector_type(4)));

__device__ __forceinline__ float wave_sum(float v) {
#pragma unroll
  for (int m = 16; m >= 1; m >>= 1) v += __shfl_xor(v, m, 32);
  return v;
}

// Generic (flat) pointer to a __shared__ object keeps the LDS byte offset in
// its low 32 bits (ISA 10.2, LDS aperture).
__device__ __forceinline__ unsigned lds_addr32(const void* p) {
  return (unsigned)(uintptr_t)p;
}

__global__ __launch_bounds__(kThreads) void context_gating_kernel(
    const float* __restrict__ hidden,   // [T, H]
    const float* __restrict__ engrams,  // [T, K, H]
    const float* __restrict__ w_h,      // [H]
    const float* __restrict__ w_e,      // [H]
    const float* __restrict__ bias_p,   // [1]
    float* __restrict__ contrib,        // [T, H]
    float* __restrict__ alphas)         // [T, K]
{
  __shared__ __align__(16) float s_buf[2][kH];       // engram row double buffer
  __shared__ float  s_hred[kWaves];                  // hidden ssq partials
  __shared__ float2 s_kred[kK][kWaves];              // per-k (ssq, dot) partials

  const int tid   = threadIdx.x;
  const int lane  = tid & 31;
  const int wv    = tid >> 5;
  const int token = blockIdx.x;

  const float* hrow = hidden  + (size_t)token * kH;
  const float* erow = engrams + (size_t)token * kK * kH;
  float*       crow = contrib + (size_t)token * kH;
  float*       arow = alphas  + (size_t)token * kK;

  // Per-thread 32-byte slot (two b128 transfers) within each engram row.
  const unsigned voff = (unsigned)(tid * kEPT * (int)sizeof(float));

  // --- Prologue: async prefetch of engram rows 0 and 1 into LDS ------------
  {
    const unsigned l0 = lds_addr32(&s_buf[0][tid * kEPT]);
    const unsigned l1 = lds_addr32(&s_buf[1][tid * kEPT]);
    const unsigned long long g0 = (unsigned long long)(uintptr_t)(erow);
    const unsigned long long g1 = (unsigned long long)(uintptr_t)(erow + kH);
    asm volatile(
        "global_load_async_to_lds_b128 %0, %2, %3 th:TH_LOAD_NT\n\t"
        "global_load_async_to_lds_b128 %0, %2, %3 offset:16 th:TH_LOAD_NT\n\t"
        "global_load_async_to_lds_b128 %1, %2, %4 th:TH_LOAD_NT\n\t"
        "global_load_async_to_lds_b128 %1, %2, %4 offset:16 th:TH_LOAD_NT"
        :
        : "v"(l0), "v"(l1), "v"(voff), "s"(g0), "s"(g1)
        : "memory");
  }

  // --- Hidden row RMS factor (overlaps with async prefetch) ----------------
  // Hidden rows are single-touch: non-temporal loads keep them out of L2.
  const v4f* h4 = reinterpret_cast<const v4f*>(hrow) + tid * 2;
  const v4f  xa = __builtin_nontemporal_load(&h4[0]);
  const v4f  xb = __builtin_nontemporal_load(&h4[1]);
  // Weights are reused by every block: regular-temporal (stay L2 resident).
  const v4f* wh4 = reinterpret_cast<const v4f*>(w_h) + tid * 2;
  const v4f  wha = wh4[0], whb = wh4[1];
  const v4f* we4 = reinterpret_cast<const v4f*>(w_e) + tid * 2;
  const v4f  wea = we4[0], web = we4[1];

  float x[kEPT]   = {xa.x, xa.y, xa.z, xa.w, xb.x, xb.y, xb.z, xb.w};
  float wwh[kEPT] = {wha.x, wha.y, wha.z, wha.w, whb.x, whb.y, whb.z, whb.w};
  float wwe[kEPT] = {wea.x, wea.y, wea.z, wea.w, web.x, web.y, web.z, web.w};

  float ssq = 0.f;
#pragma unroll
  for (int j = 0; j < kEPT; ++j) ssq += x[j] * x[j];
  ssq = wave_sum(ssq);
  if (lane == 0) s_hred[wv] = ssq;
  __syncthreads();
  float tot = 0.f;
#pragma unroll
  for (int w = 0; w < kWaves; ++w) tot += s_hred[w];
  const float inv_h = 1.f / sqrtf(tot * (1.f / (float)kH) + kEps);

  // g[j] = h_norm[j] * w_e[j]  (so dot(h_norm, e_norm) = inv_e * sum g*e)
  float g[kEPT];
#pragma unroll
  for (int j = 0; j < kEPT; ++j) g[j] = x[j] * inv_h * wwh[j] * wwe[j];

  const float bias = *bias_p;
  float acc[kEPT];
#pragma unroll
  for (int j = 0; j < kEPT; ++j) acc[j] = 0.f;

  // --- K loop: double-buffered async rows ----------------------------------
#pragma unroll
  for (int k = 0; k < kK; ++k) {
    // Retire row k (rows k and k+1 in flight = 4 outstanding per wave).
    if (k == kK - 1) asm volatile("s_wait_asynccnt 0" ::: "memory");
    else             asm volatile("s_wait_asynccnt 2" ::: "memory");

    const v4f* b4 = reinterpret_cast<const v4f*>(&s_buf[k & 1][tid * kEPT]);
    const v4f  ea = b4[0], eb = b4[1];
    const float e[kEPT] = {ea.x, ea.y, ea.z, ea.w, eb.x, eb.y, eb.z, eb.w};

    // Re-issue this buffer slot for row k+2. s_wait_dscnt guards the just-
    // issued DS reads of this slot against the incoming async LDS write.
    if (k + 2 < kK) {
      const unsigned l = lds_addr32(&s_buf[k & 1][tid * kEPT]);
      const unsigned long long gg =
          (unsigned long long)(uintptr_t)(erow + (size_t)(k + 2) * kH);
      asm volatile(
          "s_wait_dscnt 0\n\t"
          "global_load_async_to_lds_b128 %0, %1, %2 th:TH_LOAD_NT\n\t"
          "global_load_async_to_lds_b128 %0, %1, %2 offset:16 th:TH_LOAD_NT"
          :
          : "v"(l), "v"(voff), "s"(gg)
          : "memory");
    }

    float s2 = 0.f, d = 0.f;
#pragma unroll
    for (int j = 0; j < kEPT; ++j) {
      s2 += e[j] * e[j];
      d  += g[j] * e[j];
    }
    s2 = wave_sum(s2);
    d  = wave_sum(d);
    if (lane == 0) s_kred[k][wv] = make_float2(s2, d);
    __syncthreads();

    float S2 = 0.f, D = 0.f;
#pragma unroll
    for (int w = 0; w < kWaves; ++w) {
      const float2 t = s_kred[k][w];
      S2 += t.x;
      D  += t.y;
    }
    const float inv_e = 1.f / sqrtf(S2 * (1.f / (float)kH) + kEps);
    const float z     = D * inv_e * kScale + bias;
    const float alpha = 1.f / (1.f + expf(-z));
    if (tid == 0) arow[k] = alpha;

#pragma unroll
    for (int j = 0; j < kEPT; ++j) acc[j] += alpha * e[j];
  }

  // --- Store contribution row (single-touch: non-temporal) -----------------
  v4f* c4 = reinterpret_cast<v4f*>(crow) + tid * 2;
  const v4f oa = {acc[0], acc[1], acc[2], acc[3]};
  const v4f ob = {acc[4], acc[5], acc[6], acc[7]};
  __builtin_nontemporal_store(oa, &c4[0]);
  __builtin_nontemporal_store(ob, &c4[1]);
}

extern "C" void kernel_launch(void* const* d_in, const int* in_sizes, int n_in,
                              void* d_out, int out_size, void* d_ws, size_t ws_size,
                              hipStream_t stream) {
  const float* hidden  = (const float*)d_in[0];  // [B,S,H]
  const float* engrams = (const float*)d_in[1];  // [B,S,K,H]
  const float* w_h     = (const float*)d_in[2];  // [H]
  const float* w_e     = (const float*)d_in[3];  // [H]
  const float* bias    = (const float*)d_in[4];  // scalar

  const int tokens = in_sizes[0] / kH;           // B*S = 4096
  float* contrib = (float*)d_out;                       // [T,H] first
  float* alphas  = (float*)d_out + (size_t)tokens * kH; // then [T,K]

  hipLaunchKernelGGL(context_gating_kernel, dim3(tokens), dim3(kThreads), 0,
                     stream, hidden, engrams, w_h, w_e, bias, contrib, alphas);
}